// RNNTLoss_4011499454873
// MI455X (gfx1250) — compile-verified
//
#include <hip/hip_runtime.h>
#include <math.h>

// Problem constants (from reference)
#define Bn 8
#define Tn 128
#define Un 64
#define U1 (Un + 1)
#define Vn 1024
#define BLANK (Vn - 1)

typedef __attribute__((ext_vector_type(2))) float v2f;
typedef __attribute__((ext_vector_type(4))) float v4f;
typedef __attribute__((ext_vector_type(8))) float v8f;

// -------------------------------------------------------------------------
// Kernel 1: per-(b,t,u) row logsumexp over V=1024, streaming memory-bound.
// One wave32 per row. Coalesced float4 loads, values kept in registers.
// Cross-lane sum-of-exp reduction done on the matrix pipe with
// V_WMMA_F32_16X16X4_F32 (A = per-lane partial sums, B = ones).
// -------------------------------------------------------------------------
__global__ __launch_bounds__(256) void rnnt_lse_kernel(
    const float* __restrict__ logits,   // (B, T, U+1, V)
    const int* __restrict__ targets,    // (B, U)
    float* __restrict__ lpb,            // (B, T, U+1)  = lp[..., blank]
    float* __restrict__ lpe)            // (B, T, U)    = lp at target label
{
    const int wave = threadIdx.x >> 5;
    const int lane = threadIdx.x & 31;
    const int row  = blockIdx.x * 8 + wave;      // row in [0, B*T*(U+1))

    const float* rp = logits + (size_t)row * Vn;

    // ---- coalesced vectorized load: lane takes float4 #(lane + 32*j) ----
    float vals[32];
    const v4f* rp4 = (const v4f*)rp;
#pragma unroll
    for (int j = 0; j < 8; ++j) {
        v4f v = rp4[lane + 32 * j];
        vals[4 * j + 0] = v.x;
        vals[4 * j + 1] = v.y;
        vals[4 * j + 2] = v.z;
        vals[4 * j + 3] = v.w;
    }

    // ---- wave max (online-softmax pivot) ----
    float m = vals[0];
#pragma unroll
    for (int i = 1; i < 32; ++i) m = fmaxf(m, vals[i]);
#pragma unroll
    for (int off = 16; off >= 1; off >>= 1)
        m = fmaxf(m, __shfl_xor(m, off, 32));

    // ---- per-lane partial sums of exp(x - max), split in two halves ----
    float s0 = 0.f, s1 = 0.f;
#pragma unroll
    for (int i = 0; i < 16; ++i) s0 += __expf(vals[i] - m);
#pragma unroll
    for (int i = 16; i < 32; ++i) s1 += __expf(vals[i] - m);

    // ---- matrix-pipe reduction: A(16x4 f32, 2 VGPRs) x ones(4x16) ----
    // A[m][k]: lanes 0-15 hold (k=0,k=1), lanes 16-31 hold (k=2,k=3).
    // With B = ones, D[m][n] = rowsum(A[m]); summing D's 8 VGPRs per lane
    // and adding the xor-16 partner yields sum of all 64 A entries
    // (= sum over all 32 lanes of s0+s1) broadcast to every lane.
    v2f a; a.x = s0; a.y = s1;
    v2f bones; bones.x = 1.0f; bones.y = 1.0f;
    v8f c = {};
    c = __builtin_amdgcn_wmma_f32_16x16x4_f32(
        /*neg_a=*/false, a, /*neg_b=*/false, bones,
        /*c_mod=*/(short)0, c, /*reuse_a=*/false, /*reuse_b=*/false);

    float tsum = c[0] + c[1] + c[2] + c[3] + c[4] + c[5] + c[6] + c[7];
    tsum += __shfl_xor(tsum, 16, 32);

    const float lse = m + __logf(tsum);

    if (lane == 0) {
        const int u  = row % U1;
        const int bt = row / U1;        // = b*T + t
        const int bb = bt / Tn;
        lpb[row] = rp[BLANK] - lse;     // scalar re-read hits cache
        if (u < Un) {
            const int tgt = targets[bb * Un + u];
            lpe[bt * Un + u] = rp[tgt] - lse;
        }
    }
}

// -------------------------------------------------------------------------
// Kernel 2: RNNT forward-variable DP, one block per batch element.
// alpha kept in LDS (128*65*4B = 33 KB << 320 KB/WGP); anti-diagonal
// wavefront: cell (t,u) depends on (t-1,u) and (t,u-1), both on the
// previous diagonal. 192 diagonals, barrier per diagonal.
// Recurrence (matches reference incl. t=0 cumsum row and u=0 column):
//   alpha[t][u] = logaddexp( alpha[t-1][u] + lpb[t-1][u],
//                            alpha[t][u-1] + lpe[t][u-1] )
// with missing terms treated as -inf and alpha[0][0] = 0.
// -------------------------------------------------------------------------
__global__ __launch_bounds__(128) void rnnt_dp_kernel(
    const float* __restrict__ lpb,            // (B, T, U+1)
    const float* __restrict__ lpe,            // (B, T, U)
    const int* __restrict__ logit_lengths,    // (B,)
    const int* __restrict__ target_lengths,   // (B,)
    float* __restrict__ out)                  // (B,)
{
    __shared__ float alpha[Tn * U1];

    const int b   = blockIdx.x;
    const int tid = threadIdx.x;
    const float* pb = lpb + (size_t)b * Tn * U1;
    const float* pe = lpe + (size_t)b * Tn * Un;
    const float NEG_INF = -INFINITY;

    for (int d = 0; d < Tn + Un; ++d) {
        const int u = tid;
        if (u <= Un) {
            const int t = d - u;
            if (t >= 0 && t < Tn) {
                float up = (t > 0) ? alpha[(t - 1) * U1 + u] + pb[(t - 1) * U1 + u]
                                   : NEG_INF;
                float lf = (u > 0) ? alpha[t * U1 + (u - 1)] + pe[t * Un + (u - 1)]
                                   : NEG_INF;
                float v;
                if (t == 0 && u == 0) {
                    v = 0.0f;
                } else if (up == NEG_INF) {
                    v = lf;
                } else if (lf == NEG_INF) {
                    v = up;
                } else {
                    const float mx = fmaxf(up, lf);
                    v = mx + __logf(__expf(up - mx) + __expf(lf - mx));
                }
                alpha[t * U1 + u] = v;
            }
        }
        __syncthreads();
    }

    if (tid == 0) {
        const int tl = logit_lengths[b] - 1;
        const int ul = target_lengths[b];
        out[b] = -(alpha[tl * U1 + ul] + pb[tl * U1 + ul]);
    }
}

// -------------------------------------------------------------------------
// Launcher
// -------------------------------------------------------------------------
extern "C" void kernel_launch(void* const* d_in, const int* in_sizes, int n_in,
                              void* d_out, int out_size, void* d_ws, size_t ws_size,
                              hipStream_t stream) {
    (void)in_sizes; (void)n_in; (void)out_size; (void)ws_size;

    const float* logits         = (const float*)d_in[0];
    const int*   targets        = (const int*)d_in[1];
    const int*   logit_lengths  = (const int*)d_in[2];
    const int*   target_lengths = (const int*)d_in[3];
    float*       out            = (float*)d_out;

    // workspace layout: lpb (B*T*(U+1) f32) then lpe (B*T*U f32) ~ 516 KB
    float* lpb = (float*)d_ws;
    float* lpe = lpb + (size_t)Bn * Tn * U1;

    const int rows = Bn * Tn * U1;            // 66,560 rows, 8 waves/block
    rnnt_lse_kernel<<<rows / 8, 256, 0, stream>>>(logits, targets, lpb, lpe);
    rnnt_dp_kernel<<<Bn, 128, 0, stream>>>(lpb, lpe, logit_lengths,
                                           target_lengths, out);
}